// ConvEnhanced_65481071405356
// MI455X (gfx1250) — compile-verified
//
#include <hip/hip_runtime.h>
#include <hip/hip_bf16.h>
#include <math.h>

typedef __attribute__((ext_vector_type(2))) float v2f;
typedef __attribute__((ext_vector_type(8))) float v8f;

// One wave32 computes everything. The four depthwise dot products (each only
// 4 terms, since the k=2 crop zero-pads all later kernels) are evaluated with
// a single V_WMMA_F32_16X16X4_F32:  A rows 0..3 = (d00,d01,d10,d11),
// B column i = kernel-i's weights at the surviving 2x2 positions,
// so D[i][i] = dot_i.
__global__ __launch_bounds__(32) void ConvEnhanced_65481071405356_kernel(
    const float* __restrict__ data, const float* __restrict__ dw2,
    const float* __restrict__ dw3, const float* __restrict__ dw5,
    const float* __restrict__ dw7, const float* __restrict__ pw_w,
    const float* __restrict__ pw_b, const float* __restrict__ attn_w,
    float* __restrict__ out)
{
    const int lane = threadIdx.x;

    // Cropped 2x2 of the original 8x8 (row-major): data[0],data[1],data[8],data[9]
    const float d00 = data[0], d01 = data[1], d10 = data[8], d11 = data[9];

    // --- Build WMMA fragments (ISA 16x4 f32 A layout / mirrored B layout) ---
    // lanes 0-15  : VGPR0=K0, VGPR1=K1   (A: M=lane, B: N=lane)
    // lanes 16-31 : VGPR0=K2, VGPR1=K3   (A: M=lane-16, B: N=lane-16)
    v2f a = {0.0f, 0.0f};
    v2f b = {0.0f, 0.0f};
    {
        const int half = lane >> 4;
        const int idx  = lane & 15;
        if (idx < 4) {
            a.x = half ? d10 : d00;
            a.y = half ? d11 : d01;
            const float* w; int s;
            switch (idx) {
                case 0:  w = dw2; s = 2; break;
                case 1:  w = dw3; s = 3; break;
                case 2:  w = dw5; s = 5; break;
                default: w = dw7; s = 7; break;
            }
            // weights at flat positions [0],[1],[s],[s+1] (top-left 2x2)
            b.x = half ? w[s]     : w[0];
            b.y = half ? w[s + 1] : w[1];
        }
    }

    v8f c = {};
    // 8 args: (neg_a, A, neg_b, B, c_mod, C, reuse_a, reuse_b)
    c = __builtin_amdgcn_wmma_f32_16x16x4_f32(
        false, a, false, b, (short)0, c, false, false);

    // D[i][i] lives in VGPR i of lane i (lanes 0..3 are in the 0-15 half).
    const float dot0 = __shfl(c[0], 0, 32);
    const float dot1 = __shfl(c[1], 1, 32);
    const float dot2 = __shfl(c[2], 2, 32);
    const float dot3 = __shfl(c[3], 3, 32);

    if (lane == 0) {
        const float dots[4] = {dot0, dot1, dot2, dot3};
        const int   ks[4]   = {2, 3, 5, 7};

        float outs_sum  = 0.0f;
        float attns_sum = 0.0f;
        for (int i = 0; i < 4; ++i) {
            // pointwise: scale + bias, then sigmoid (mean of 1x1 == itself)
            const float logit = dots[i] * pw_w[i] + pw_b[i];
            outs_sum += 1.0f / (1.0f + expf(-logit));

            // mean(softmax) over k*k logits: 4 nonzero (data * attn_w[i]),
            // the remaining k*k-4 are exactly 0 (zero-padded data).
            const int   n  = ks[i] * ks[i];
            const float aw = attn_w[i];
            const float l0 = d00 * aw, l1 = d01 * aw;
            const float l2 = d10 * aw, l3 = d11 * aw;
            float mx = fmaxf(fmaxf(l0, l1), fmaxf(l2, l3));
            if (n > 4) mx = fmaxf(mx, 0.0f);
            const float e0 = expf(l0 - mx), e1 = expf(l1 - mx);
            const float e2 = expf(l2 - mx), e3 = expf(l3 - mx);
            const float ez = (n > 4) ? expf(0.0f - mx) : 0.0f;
            const float s  = e0 + e1 + e2 + e3 + (float)(n - 4) * ez;
            const float softmax_sum =
                e0 / s + e1 / s + e2 / s + e3 / s + (float)(n - 4) * (ez / s);
            attns_sum += softmax_sum / (float)n;
        }

        const float mean_out = outs_sum * 0.25f;
        const float resid    = (d00 + d01 + d10 + d11) / 49.0f; // final 7x7 mean
        out[0] = (mean_out + resid) * (attns_sum * 0.25f);
    }
}

extern "C" void kernel_launch(void* const* d_in, const int* in_sizes, int n_in,
                              void* d_out, int out_size, void* d_ws, size_t ws_size,
                              hipStream_t stream) {
    (void)in_sizes; (void)n_in; (void)out_size; (void)d_ws; (void)ws_size;
    ConvEnhanced_65481071405356_kernel<<<1, 32, 0, stream>>>(
        (const float*)d_in[0],  // data  (8x8)
        (const float*)d_in[1],  // dw2   (2x2)
        (const float*)d_in[2],  // dw3   (3x3)
        (const float*)d_in[3],  // dw5   (5x5)
        (const float*)d_in[4],  // dw7   (7x7)
        (const float*)d_in[5],  // pw_w  (4)
        (const float*)d_in[6],  // pw_b  (4)
        (const float*)d_in[7],  // attn_w(4)
        (float*)d_out);
}